// Denoise_17669495455833
// MI455X (gfx1250) — compile-verified
//
#include <hip/hip_runtime.h>

typedef float v2f __attribute__((ext_vector_type(2)));
typedef float v8f __attribute__((ext_vector_type(8)));

namespace {

constexpr float kLam   = 10.0f;
constexpr int   kIters = 100;   // per pass; reference runs two passes
constexpr int   kRows  = 16;    // rows per block == WMMA M
constexpr int   kN     = 512;   // row length
// LDS row stride in floats: [2 zero pad][512 z][2 zero pad][8 slack]
// 8*524 mod 64 == 32 -> the M and M+8 fragment halves hit disjoint banks.
constexpr int   kStride        = 524;
constexpr int   kWaves         = 8;
constexpr int   kTilesPerWave  = 4;   // 512/16 tiles / 8 waves

// Coefficient of (D^T D) as a 20x16 window matrix per output tile.
// Window row k (0..19) corresponds to ze[j0+k]  (z index j0+k-2; ze has a
// 2-column zero pad on each side).  Output column n corresponds to z[j0+n].
// kind: 0 = leftmost tile (j0==0), 1 = interior, 2 = rightmost (j0==496).
__device__ __forceinline__ float dtd_coef(int k, int n, int kind) {
  if (kind == 0 && n < 2) {
    if (n == 0) {                     // row 0 of D^T D: [1,-2,1]
      if (k == 2) return  1.f;
      if (k == 3) return -2.f;
      if (k == 4) return  1.f;
      return 0.f;
    } else {                          // row 1: [-2,5,-4,1]
      if (k == 2) return -2.f;
      if (k == 3) return  5.f;
      if (k == 4) return -4.f;
      if (k == 5) return  1.f;
      return 0.f;
    }
  }
  if (kind == 2 && n >= 14) {
    if (n == 14) {                    // row n-2: [1,-4,5,-2]
      if (k == 14) return  1.f;
      if (k == 15) return -4.f;
      if (k == 16) return  5.f;
      if (k == 17) return -2.f;
      return 0.f;
    } else {                          // row n-1: [1,-2,1]
      if (k == 15) return  1.f;
      if (k == 16) return -2.f;
      if (k == 17) return  1.f;
      return 0.f;
    }
  }
  int t = k - n;                      // interior: Toeplitz [1,-4,6,-4,1]
  float r = 0.f;
  if (t == 0 || t == 4) r = 1.f;
  else if (t == 1 || t == 3) r = -4.f;
  else if (t == 2) r = 6.f;
  return r;
}

__global__ __launch_bounds__(256) void fista_wmma_kernel(
    const float* __restrict__ y_in, float* __restrict__ out) {
  __shared__ float ze[kRows][kStride];

  const int lane = threadIdx.x & 31;
  const int wave = threadIdx.x >> 5;
  const int mlo  = lane & 15;   // N column / A-row index
  const int hi   = lane >> 4;   // 0 or 1 (lane half)
  const int r0   = blockIdx.x * kRows;   // global row base of this band

  // FISTA state in C/D fragment layout: element (q, g) of this lane is
  // matrix element M = g + 8*hi, N = mlo of tile q (j0 = (wave*4+q)*16).
  float yf[kTilesPerWave][8];
  float xf[kTilesPerWave][8];
  float zf[kTilesPerWave][8];

#pragma unroll
  for (int q = 0; q < kTilesPerWave; ++q) {
    const int j0 = (wave * kTilesPerWave + q) * 16;
#pragma unroll
    for (int g = 0; g < 8; ++g) {
      const int M = g + hi * 8;
      float v = y_in[(size_t)(r0 + M) * kN + (j0 + mlo)];
      yf[q][g] = v;
      float x0 = fminf(fmaxf(v, 0.f), v);   // proj(y) onto [0,y]
      xf[q][g] = x0;
      zf[q][g] = x0;
    }
  }

  // Zero the pad columns once (z writes never touch them).
  for (int i = threadIdx.x; i < kRows * kStride; i += blockDim.x) {
    const int c = i % kStride;
    if (c < 2 || c >= 2 + kN) ze[i / kStride][c] = 0.f;
  }
  // Initial z image.
#pragma unroll
  for (int q = 0; q < kTilesPerWave; ++q) {
    const int j0 = (wave * kTilesPerWave + q) * 16;
#pragma unroll
    for (int g = 0; g < 8; ++g) ze[g + hi * 8][2 + j0 + mlo] = zf[q][g];
  }

  // Per-tile B fragments (20x16 stencil matrix in five K=4 chunks).
  // B layout assumption: reg r of lane l holds S[4*ch + 2*hi + r][mlo].
  v2f Btile[kTilesPerWave][5];
#pragma unroll
  for (int q = 0; q < kTilesPerWave; ++q) {
    const int j0   = (wave * kTilesPerWave + q) * 16;
    const int kind = (j0 == 0) ? 0 : ((j0 == kN - 16) ? 2 : 1);
#pragma unroll
    for (int ch = 0; ch < 5; ++ch) {
      v2f b;
      b[0] = dtd_coef(4 * ch + 2 * hi + 0, mlo, kind);
      b[1] = dtd_coef(4 * ch + 2 * hi + 1, mlo, kind);
      Btile[q][ch] = b;
    }
  }
  __syncthreads();

  const float step = 1.0f / (2.0f * (1.0f + 16.0f * kLam));
  float t = 1.0f;

  for (int pass = 0; pass < 2; ++pass) {
    if (pass == 1) {
      // Restart: y <- pass-1 result, x0 = z0 = proj(y) = y (all >= 0), t = 1.
      t = 1.0f;
#pragma unroll
      for (int q = 0; q < kTilesPerWave; ++q) {
        const int j0 = (wave * kTilesPerWave + q) * 16;
#pragma unroll
        for (int g = 0; g < 8; ++g) {
          yf[q][g] = xf[q][g];
          zf[q][g] = xf[q][g];
          ze[g + hi * 8][2 + j0 + mlo] = zf[q][g];
        }
      }
      __syncthreads();
    }

    for (int it = 0; it < kIters; ++it) {
      const float tn  = 0.5f * (1.0f + sqrtf(1.0f + 4.0f * t * t));
      const float mom = (t - 1.0f) / tn;

      // Stage 1: dtd = (D^T D) z per tile via full-FP32 WMMA.
      v8f acc[kTilesPerWave];
#pragma unroll
      for (int q = 0; q < kTilesPerWave; ++q) {
        const int j0 = (wave * kTilesPerWave + q) * 16;
        v8f c = {};
#pragma unroll
        for (int ch = 0; ch < 5; ++ch) {
          // A fragment: lane holds row M=mlo, K = 4*ch + 2*hi + {0,1} of the
          // window ze[j0 .. j0+19] (8-byte aligned contiguous pair).
          const v2f a =
              *reinterpret_cast<const v2f*>(&ze[mlo][j0 + 4 * ch + 2 * hi]);
          c = __builtin_amdgcn_wmma_f32_16x16x4_f32(
              false, a, false, Btile[q][ch], (short)0, c, false, false);
        }
        acc[q] = c;
      }
      __syncthreads();   // all waves done reading ze

      // Stage 2: elementwise FISTA update, publish z_new.
#pragma unroll
      for (int q = 0; q < kTilesPerWave; ++q) {
        const int j0 = (wave * kTilesPerWave + q) * 16;
#pragma unroll
        for (int g = 0; g < 8; ++g) {
          const float z = zf[q][g];
          const float y = yf[q][g];
          const float grad = 2.0f * (z - y) + (2.0f * kLam) * acc[q][g];
          float xn = z - step * grad;
          xn = fminf(fmaxf(xn, 0.f), y);           // proj onto [0, y]
          const float zn = xn + mom * (xn - xf[q][g]);
          xf[q][g] = xn;
          zf[q][g] = zn;
          ze[g + hi * 8][2 + j0 + mlo] = zn;
        }
      }
      __syncthreads();
      t = tn;
    }
  }

  // Write final x (flat layout of (32,512,512,1) == (32,512,512)).
#pragma unroll
  for (int q = 0; q < kTilesPerWave; ++q) {
    const int j0 = (wave * kTilesPerWave + q) * 16;
#pragma unroll
    for (int g = 0; g < 8; ++g) {
      const int M = g + hi * 8;
      out[(size_t)(r0 + M) * kN + (j0 + mlo)] = xf[q][g];
    }
  }
}

}  // namespace

extern "C" void kernel_launch(void* const* d_in, const int* in_sizes, int n_in,
                              void* d_out, int out_size, void* d_ws, size_t ws_size,
                              hipStream_t stream) {
  const float* data = (const float*)d_in[0];
  float* out = (float*)d_out;
  const int total  = in_sizes[0];        // 32*512*512
  const int rows   = total / kN;         // 16384
  const int blocks = rows / kRows;       // 1024 workgroups of 16 rows
  hipLaunchKernelGGL(fista_wmma_kernel, dim3(blocks), dim3(kWaves * 32), 0,
                     stream, data, out);
}